// DeepSupervisionLoss_63496796504196
// MI455X (gfx1250) — compile-verified
//
#include <hip/hip_runtime.h>
#include <hip/hip_bf16.h>

// DeepSupervisionLoss on MI455X (gfx1250, wave32).
// Bandwidth-bound streaming reduction: 168 MB read @ 23.3 TB/s ~= 7.2 us floor;
// ~16 hardware transcendentals/pixel (exp/log/rcp/sqrt x 4 heads) is roughly
// balanced against that, so the loop is vectorized x4 (b128 loads) and fully
// branchless (3x3 clamp-to-edge == in-bounds-only max/min for morphology).
// Pass 1: grid-stride float4 loop, 13 partial sums per thread, exact 32-lane
//         reduction via V_WMMA_F32_16X16X4_F32 (A = partials, B = ones).
// Pass 2: single block combines block partials, evaluates the scalar.

typedef __attribute__((ext_vector_type(2))) float v2f;
typedef __attribute__((ext_vector_type(8))) float v8f;

#define TOTAL_ELEMS (32 * 512 * 512)
#define TOTAL4 (TOTAL_ELEMS / 4)
#define NSUMS 13  // [0..3]=focal, [4..7]=intersection, [8..11]=pred_sum, [12]=target_sum

// Exact 32-lane sum using the f32 WMMA:
// A(16x4): A[m][0]=p_m (lanes 0-15, V0), A[m][2]=p_{m+16} (lanes 16-31, V0), V1=0.
// B = ones -> D[m][n] = p_m + p_{m+16} (replicated over n).
// Sum D's 8 VGPRs: lanes<16 get rows 0-7, lanes>=16 get rows 8-15; xor-16 combines.
__device__ __forceinline__ float wave_reduce_f32(float p) {
  v2f a; a.x = p;    a.y = 0.0f;
  v2f b; b.x = 1.0f; b.y = 1.0f;
  v8f c = {};
  v8f d = __builtin_amdgcn_wmma_f32_16x16x4_f32(
      /*neg_a=*/false, a, /*neg_b=*/false, b,
      /*c_mod=*/(short)0, c, /*reuse_a=*/false, /*reuse_b=*/false);
  float s = ((d[0] + d[1]) + (d[2] + d[3])) + ((d[4] + d[5]) + (d[6] + d[7]));
  s += __shfl_xor(s, 16);
  return s;
}

__global__ __launch_bounds__(256) void dsl_partials_kernel(
    const float* __restrict__ p0, const float* __restrict__ p1,
    const float* __restrict__ p2, const float* __restrict__ p3,
    const float* __restrict__ tgt, float* __restrict__ ws, int nblocks) {
  float accF0 = 0.f, accF1 = 0.f, accF2 = 0.f, accF3 = 0.f;
  float accI0 = 0.f, accI1 = 0.f, accI2 = 0.f, accI3 = 0.f;
  float accP0 = 0.f, accP1 = 0.f, accP2 = 0.f, accP3 = 0.f;
  float accT = 0.f;

  const float4* q0 = reinterpret_cast<const float4*>(p0);
  const float4* q1 = reinterpret_cast<const float4*>(p1);
  const float4* q2 = reinterpret_cast<const float4*>(p2);
  const float4* q3 = reinterpret_cast<const float4*>(p3);
  const float4* t4 = reinterpret_cast<const float4*>(tgt);

  const int stride = gridDim.x * blockDim.x;
  for (int g = blockIdx.x * blockDim.x + threadIdx.x; g < TOTAL4; g += stride) {
    // g indexes float4s: 128 per row, 512 rows per image.
    const int x4 = g & 127;
    const int y = (g >> 7) & 511;
    const int imgbase4 = (g >> 16) << 16;     // float4 index of image start
    const int imgbase = imgbase4 << 2;        // float index of image start
    const int x0 = x4 << 2;

    // clamp-to-edge == SAME-window max/min with OOB excluded (duplicates only)
    const int yym = (y > 0) ? y - 1 : 0;
    const int yyp = (y < 511) ? y + 1 : 511;
    const int xl = (x0 > 0) ? x0 - 1 : 0;
    const int xr = (x0 + 4 < 512) ? x0 + 4 : 511;

    const float4 r0 = t4[imgbase4 + yym * 128 + x4];
    const float4 r1 = t4[imgbase4 + y   * 128 + x4];
    const float4 r2 = t4[imgbase4 + yyp * 128 + x4];
    const float l0 = tgt[imgbase + yym * 512 + xl], e0 = tgt[imgbase + yym * 512 + xr];
    const float l1 = tgt[imgbase + y   * 512 + xl], e1 = tgt[imgbase + y   * 512 + xr];
    const float l2 = tgt[imgbase + yyp * 512 + xl], e2 = tgt[imgbase + yyp * 512 + xr];

    // separable morphology: vertical max/min over the 6 columns [l, x, y, z, w, r]
    float vmx[6], vmn[6];
    vmx[0] = fmaxf(fmaxf(l0, l1), l2);          vmn[0] = fminf(fminf(l0, l1), l2);
    vmx[1] = fmaxf(fmaxf(r0.x, r1.x), r2.x);    vmn[1] = fminf(fminf(r0.x, r1.x), r2.x);
    vmx[2] = fmaxf(fmaxf(r0.y, r1.y), r2.y);    vmn[2] = fminf(fminf(r0.y, r1.y), r2.y);
    vmx[3] = fmaxf(fmaxf(r0.z, r1.z), r2.z);    vmn[3] = fminf(fminf(r0.z, r1.z), r2.z);
    vmx[4] = fmaxf(fmaxf(r0.w, r1.w), r2.w);    vmn[4] = fminf(fminf(r0.w, r1.w), r2.w);
    vmx[5] = fmaxf(fmaxf(e0, e1), e2);          vmn[5] = fminf(fminf(e0, e1), e2);

    const float4 a0 = q0[g], a1 = q1[g], a2 = q2[g], a3 = q3[g];
    const float cen[4] = {r1.x, r1.y, r1.z, r1.w};
    const float xs0[4] = {a0.x, a0.y, a0.z, a0.w};
    const float xs1[4] = {a1.x, a1.y, a1.z, a1.w};
    const float xs2[4] = {a2.x, a2.y, a2.z, a2.w};
    const float xs3[4] = {a3.x, a3.y, a3.z, a3.w};

#pragma unroll
    for (int i = 0; i < 4; ++i) {
      const float mx = fmaxf(fmaxf(vmx[i], vmx[i + 1]), vmx[i + 2]);
      const float mn = fminf(fminf(vmn[i], vmn[i + 1]), vmn[i + 2]);
      const float t = cen[i];
      const float w = 1.0f + 7.0f * (mx - mn);  // 1 + boundary*(8-1)
      accT += t * w;
      const float tw = t * w;

      const float xs[4] = {xs0[i], xs1[i], xs2[i], xs3[i]};
      float fo[4], in[4], ps[4];
#pragma unroll
      for (int r = 0; r < 4; ++r) {
        const float xv = xs[r];
        const float ax = fabsf(xv);
        const float e = __expf(-ax);            // exp(-|x|)
        const float l1p = __logf(1.0f + e);     // log1p(exp(-|x|))
        const float bce = fmaxf(xv, 0.0f) - xv * t + l1p;
        const float inv = __builtin_amdgcn_rcpf(1.0f + e);
        const float p = (xv >= 0.0f) ? inv : e * inv;    // sigmoid(x)
        const float omp = (t == 1.0f) ? (1.0f - p) : p;  // 1 - pt
        // 0.25 * omp^2.5 == 0.25 * omp*omp*sqrt(omp)  (exact for gamma=2.5)
        const float fw = 0.25f * omp * omp * __builtin_amdgcn_sqrtf(omp);
        fo[r] = fw * bce;
        in[r] = p * tw;
        ps[r] = p * w;
      }
      accF0 += fo[0]; accF1 += fo[1]; accF2 += fo[2]; accF3 += fo[3];
      accI0 += in[0]; accI1 += in[1]; accI2 += in[2]; accI3 += in[3];
      accP0 += ps[0]; accP1 += ps[1]; accP2 += ps[2]; accP3 += ps[3];
    }
  }

  float acc[NSUMS] = {accF0, accF1, accF2, accF3,
                      accI0, accI1, accI2, accI3,
                      accP0, accP1, accP2, accP3, accT};

  __shared__ float sdata[NSUMS][8];
  const int lane = threadIdx.x & 31;
  const int wid = threadIdx.x >> 5;
#pragma unroll
  for (int j = 0; j < NSUMS; ++j) {
    const float tot = wave_reduce_f32(acc[j]);  // v_wmma_f32_16x16x4_f32
    if (lane == 0) sdata[j][wid] = tot;
  }
  __syncthreads();
  if (threadIdx.x < NSUMS) {
    float s = 0.f;
#pragma unroll
    for (int wv = 0; wv < 8; ++wv) s += sdata[threadIdx.x][wv];
    ws[threadIdx.x * nblocks + blockIdx.x] = s;
  }
}

__global__ __launch_bounds__(256) void dsl_finalize_kernel(
    const float* __restrict__ ws, int nblocks, float* __restrict__ out) {
  __shared__ float red[256];
  __shared__ float sums[NSUMS];
  for (int j = 0; j < NSUMS; ++j) {
    float s = 0.f;
    for (int i = threadIdx.x; i < nblocks; i += 256) s += ws[j * nblocks + i];
    red[threadIdx.x] = s;
    __syncthreads();
    for (int off = 128; off > 0; off >>= 1) {
      if (threadIdx.x < off) red[threadIdx.x] += red[threadIdx.x + off];
      __syncthreads();
    }
    if (threadIdx.x == 0) sums[j] = red[0];
    __syncthreads();
  }
  if (threadIdx.x == 0) {
    const float N = (float)TOTAL_ELEMS;
    const float T = sums[12];
    const float lw[4] = {1.0f, 0.4f, 0.2f, 0.1f};
    float total = 0.f;
#pragma unroll
    for (int r = 0; r < 4; ++r) {
      const float F = sums[r];
      const float I = sums[4 + r];
      const float P = sums[8 + r];
      const float dice_loss = 1.0f - (2.0f * I + 1.0f) / (P + T + 1.0f);
      total += lw[r] * (0.3f * (F / N) + 0.7f * dice_loss);
    }
    out[0] = total;
  }
}

extern "C" void kernel_launch(void* const* d_in, const int* in_sizes, int n_in,
                              void* d_out, int out_size, void* d_ws, size_t ws_size,
                              hipStream_t stream) {
  const float* p0 = (const float*)d_in[0];
  const float* p1 = (const float*)d_in[1];
  const float* p2 = (const float*)d_in[2];
  const float* p3 = (const float*)d_in[3];
  const float* tg = (const float*)d_in[4];
  float* ws = (float*)d_ws;

  size_t maxb = ws_size / (NSUMS * sizeof(float));
  int nblocks = 1024;
  if ((size_t)nblocks > maxb) nblocks = (int)maxb;
  if (nblocks < 1) nblocks = 1;

  dsl_partials_kernel<<<nblocks, 256, 0, stream>>>(p0, p1, p2, p3, tg, ws, nblocks);
  dsl_finalize_kernel<<<1, 256, 0, stream>>>(ws, nblocks, (float*)d_out);
}